// IEAD_72181220376623
// MI455X (gfx1250) — compile-verified
//
#include <hip/hip_runtime.h>
#include <hip/hip_bf16.h>
#include <cstddef>

#define N_NODES 100000
#define F_DIM   256
#define D_DIM   128
#define M_CH    10
#define B_BATCH 32768
#define K_NEI   32

#define T1 10   // row tiles per block in GEMM1 (6250 tiles / 10 = 625 blocks)
#define T2 8    // row tiles per block in GEMM2 (2048 tiles / 8  = 256 blocks)

typedef __attribute__((ext_vector_type(2))) float v2f;
typedef __attribute__((ext_vector_type(8))) float v8f;

// ---------------- K1: flow_emb = feature @ W_emb + bias  (WMMA f32 16x16x4)
// Block = 8 waves; wave w owns output cols [16w,16w+16). K-loop outermost:
// one B fragment per K-step is reused across T1 row tiles (10 accumulators).
// feature is streamed with non-temporal loads (read-once; keep L2 for flow_emb).
__global__ void __launch_bounds__(256) k_gemm_flow(
    const float* __restrict__ feat, const float* __restrict__ wemb,
    const float* __restrict__ bemb, float* __restrict__ flow) {
  const int lane = threadIdx.x & 31, wave = threadIdx.x >> 5;
  const int row  = lane & 15,  hi  = lane >> 4;
  const int col  = wave * 16 + row;
  const int tile0 = blockIdx.x * T1;
  v8f c[T1];
  #pragma unroll
  for (int t = 0; t < T1; ++t) c[t] = (v8f){};
  const float* a0 = feat + (size_t)(tile0 * 16 + row) * F_DIM + 2 * hi;
  #pragma unroll 2
  for (int s = 0; s < F_DIM / 4; ++s) {
    const int k = 4 * s;
    // B 4x16 fragment: VGPR0 = B[k+2*hi][col], VGPR1 = B[k+2*hi+1][col]
    v2f b;
    b.x = wemb[(size_t)(k + 2 * hi)     * D_DIM + col];
    b.y = wemb[(size_t)(k + 2 * hi + 1) * D_DIM + col];
    #pragma unroll
    for (int t = 0; t < T1; ++t) {
      // A 16x4 fragment: lane<16 -> A[row][k..k+1], lane>=16 -> A[row][k+2..k+3]
      v2f a = __builtin_nontemporal_load(
                  (const v2f*)(a0 + (size_t)t * 16 * F_DIM + k));
      c[t] = __builtin_amdgcn_wmma_f32_16x16x4_f32(
                 false, a, false, b, (short)0, c[t], false, false);
    }
  }
  const float bias = bemb[col];
  #pragma unroll
  for (int t = 0; t < T1; ++t) {
    float* orow = flow + (size_t)(tile0 + t) * 16 * D_DIM + col;
    #pragma unroll
    for (int v = 0; v < 8; ++v)              // D: VGPR v -> row m = v + 8*hi
      orow[(size_t)(v + 8 * hi) * D_DIM] = c[t][v] + bias;
  }
}

// ---------------- K2: per-block partial sums of character_emb = adj @ flow_emb
__global__ void __launch_bounds__(128) k_char_partial(
    const float* __restrict__ flow, const float* __restrict__ adj,
    float* __restrict__ part, int chunk) {
  const int d = threadIdx.x;
  const int n0 = blockIdx.x * chunk;
  float acc[M_CH];
  #pragma unroll
  for (int m = 0; m < M_CH; ++m) acc[m] = 0.f;
  const int n1 = min(n0 + chunk, N_NODES);
  for (int n = n0; n < n1; ++n) {
    const float v = flow[(size_t)n * D_DIM + d];
    #pragma unroll
    for (int m = 0; m < M_CH; ++m)
      acc[m] = fmaf(adj[(size_t)m * N_NODES + n], v, acc[m]);
  }
  float* p = part + (size_t)blockIdx.x * (M_CH * D_DIM);
  #pragma unroll
  for (int m = 0; m < M_CH; ++m) p[m * D_DIM + d] = acc[m];
}

// ---------------- K2b: deterministic fixed-order reduce over G partials
__global__ void __launch_bounds__(128) k_char_reduce(
    const float* __restrict__ part, float* __restrict__ charE, int G) {
  const int m = blockIdx.x, d = threadIdx.x;
  float s = 0.f;
  for (int g = 0; g < G; ++g)
    s += part[(size_t)g * (M_CH * D_DIM) + m * D_DIM + d];
  charE[m * D_DIM + d] = s;
}

// ---------------- K3: lat table [2][10][128], Weff = W1[:128]+W1[128:], score consts
__global__ void __launch_bounds__(256) k_tables(
    const float* __restrict__ charE, const float* __restrict__ wc,
    const float* __restrict__ attw,  const float* __restrict__ l1w,
    float* __restrict__ lat, float* __restrict__ weff, float* __restrict__ consts) {
  const int t = threadIdx.x;
  for (int o = t; o < 2 * M_CH * D_DIM; o += 256) {
    const int d = o & 127, combo = o >> 7;
    const int cat = combo / M_CH, pa = combo % M_CH;
    float acc = 0.f;
    for (int j = 0; j < D_DIM; ++j) {
      acc = fmaf(charE[cat * D_DIM + j], wc[(size_t)j * D_DIM + d], acc);
      acc = fmaf(charE[pa  * D_DIM + j], wc[(size_t)(D_DIM + j) * D_DIM + d], acc);
    }
    lat[o] = 1.f / (1.f + __expf(-acc));
  }
  for (int o = t; o < D_DIM * D_DIM; o += 256) {
    const int j = o >> 7, d = o & 127;
    weff[o] = l1w[(size_t)j * D_DIM + d] + l1w[(size_t)(j + D_DIM) * D_DIM + d];
  }
  if (t < 2) {
    float s = 0.f;
    for (int j = 0; j < D_DIM; ++j) s = fmaf(charE[t * D_DIM + j], attw[j], s);
    consts[t] = s;   // consts[0] = char0.attw, consts[1] = char1.attw
  }
}

// ---------------- K4: score_pre[n] = flow_all[n].attw + att1_b  (wave per node)
__global__ void __launch_bounds__(256) k_score(
    const float* __restrict__ flow, const float* __restrict__ attw,
    const float* __restrict__ adj,  const float* __restrict__ consts,
    const float* __restrict__ attb, float* __restrict__ score) {
  const int lane = threadIdx.x & 31, wv = threadIdx.x >> 5;
  const int n = blockIdx.x * 8 + wv;
  const float4 a = ((const float4*)(flow + (size_t)n * D_DIM))[lane];
  const float4 w = ((const float4*)attw)[lane];
  float s = a.x * w.x + a.y * w.y + a.z * w.z + a.w * w.w;
  #pragma unroll
  for (int off = 16; off; off >>= 1) s += __shfl_xor(s, off);
  if (lane == 0)
    score[n] = 0.5f * s
             + 0.5f * (adj[n] * consts[0] + adj[(size_t)N_NODES + n] * consts[1])
             + attb[0];
}

// ---------------- K5: neighbor softmax attention -> agg[B][128]  (wave per batch row)
__global__ void __launch_bounds__(256) k_attend(
    const float* __restrict__ flow, const float* __restrict__ score,
    const float* __restrict__ adj,  const float* __restrict__ charE,
    const int* __restrict__ history, float* __restrict__ agg) {
  const int lane = threadIdx.x & 31, wv = threadIdx.x >> 5;
  const int b = blockIdx.x * 8 + wv;
  const int h = history[(size_t)b * K_NEI + lane];      // lane k owns neighbor k
  float s = score[h];
  float mx = s;
  #pragma unroll
  for (int off = 16; off; off >>= 1) mx = fmaxf(mx, __shfl_xor(mx, off));
  float e = __expf(s - mx);
  float den = e;
  #pragma unroll
  for (int off = 16; off; off >>= 1) den += __shfl_xor(den, off);
  const float att = e / den;
  // scalar coefficients: ca = sum att_k*adj0[h_k], cb = sum att_k*adj1[h_k]
  float ca = att * adj[h];
  float cb = att * adj[(size_t)N_NODES + h];
  #pragma unroll
  for (int off = 16; off; off >>= 1) { ca += __shfl_xor(ca, off); cb += __shfl_xor(cb, off); }
  const float4 ch0 = ((const float4*)charE)[lane];
  const float4 ch1 = ((const float4*)(charE + D_DIM))[lane];
  float4 acc = {0.f, 0.f, 0.f, 0.f};
  for (int k = 0; k < K_NEI; ++k) {
    const float ak = __shfl(att, k);
    const int   hk = __shfl(h, k);
    const float4 r = ((const float4*)(flow + (size_t)hk * D_DIM))[lane];
    acc.x = fmaf(ak, r.x, acc.x); acc.y = fmaf(ak, r.y, acc.y);
    acc.z = fmaf(ak, r.z, acc.z); acc.w = fmaf(ak, r.w, acc.w);
  }
  float4 o;
  o.x = 0.5f * (acc.x + ca * ch0.x + cb * ch1.x);
  o.y = 0.5f * (acc.y + ca * ch0.y + cb * ch1.y);
  o.z = 0.5f * (acc.z + ca * ch0.z + cb * ch1.z);
  o.w = 0.5f * (acc.w + ca * ch0.w + cb * ch1.w);
  ((float4*)(agg + (size_t)b * D_DIM))[lane] = o;
}

// ---------------- K6: out = relu(agg @ Weff + b1), in-place on agg (WMMA f32)
// Same multi-tile structure as K1: B fragment reused across T2 accumulators.
__global__ void __launch_bounds__(256) k_gemm_out(
    float* __restrict__ agg, const float* __restrict__ weff,
    const float* __restrict__ l1b) {
  const int lane = threadIdx.x & 31, wave = threadIdx.x >> 5;
  const int row  = lane & 15,  hi  = lane >> 4;
  const int col  = wave * 16 + row;
  const int tile0 = blockIdx.x * T2;
  v8f c[T2];
  #pragma unroll
  for (int t = 0; t < T2; ++t) c[t] = (v8f){};
  const float* a0 = agg + (size_t)(tile0 * 16 + row) * D_DIM + 2 * hi;
  #pragma unroll 2
  for (int s = 0; s < D_DIM / 4; ++s) {
    const int k = 4 * s;
    v2f b;
    b.x = weff[(size_t)(k + 2 * hi)     * D_DIM + col];
    b.y = weff[(size_t)(k + 2 * hi + 1) * D_DIM + col];
    #pragma unroll
    for (int t = 0; t < T2; ++t) {
      v2f a = *(const v2f*)(a0 + (size_t)t * 16 * D_DIM + k);
      c[t] = __builtin_amdgcn_wmma_f32_16x16x4_f32(
                 false, a, false, b, (short)0, c[t], false, false);
    }
  }
  __syncthreads();   // all A-reads are in registers before any wave overwrites agg
  const float bias = l1b[col];
  #pragma unroll
  for (int t = 0; t < T2; ++t) {
    float* orow = agg + (size_t)(tile0 + t) * 16 * D_DIM + col;
    #pragma unroll
    for (int v = 0; v < 8; ++v)
      orow[(size_t)(v + 8 * hi) * D_DIM] = fmaxf(c[t][v] + bias, 0.f);
  }
}

// ---------------- K7: pred_r/pred_n dots via lat-table gather (wave per batch row)
__global__ void __launch_bounds__(256) k_pred(
    const float* __restrict__ outb, const float* __restrict__ lat,
    const int* __restrict__ cr, const int* __restrict__ cn,
    const int* __restrict__ pa, float* __restrict__ dout) {
  const int lane = threadIdx.x & 31, wv = threadIdx.x >> 5;
  const int b = blockIdx.x * 8 + wv;
  const float4 o  = ((const float4*)(outb + (size_t)b * D_DIM))[lane];
  const int r = cr[b], n = cn[b], p = pa[b];
  const float4 lr = ((const float4*)(lat + (size_t)(r * M_CH + p) * D_DIM))[lane];
  const float4 ln = ((const float4*)(lat + (size_t)(n * M_CH + p) * D_DIM))[lane];
  float pr = o.x * lr.x + o.y * lr.y + o.z * lr.z + o.w * lr.w;
  float pn = o.x * ln.x + o.y * ln.y + o.z * ln.z + o.w * ln.w;
  #pragma unroll
  for (int off = 16; off; off >>= 1) { pr += __shfl_xor(pr, off); pn += __shfl_xor(pn, off); }
  if (lane == 0) { dout[b] = pr; dout[B_BATCH + b] = pn; }
}

extern "C" void kernel_launch(void* const* d_in, const int* in_sizes, int n_in,
                              void* d_out, int out_size, void* d_ws, size_t ws_size,
                              hipStream_t stream) {
  const float* feature = (const float*)d_in[0];   // [N,F]
  const float* adj     = (const float*)d_in[1];   // [M,N]
  const int*   history = (const int*)  d_in[2];   // [B,K]
  const int*   cat_r   = (const int*)  d_in[4];   // [B]
  const int*   cat_n   = (const int*)  d_in[5];   // [B]
  const int*   pa_lvl  = (const int*)  d_in[6];   // [B]
  const float* wemb    = (const float*)d_in[7];   // [F,D]
  const float* bemb    = (const float*)d_in[8];   // [D]
  const float* wchar   = (const float*)d_in[9];   // [2D,D]
  const float* attw    = (const float*)d_in[10];  // [D,1]
  const float* attb    = (const float*)d_in[11];  // [1]
  const float* l1w     = (const float*)d_in[12];  // [2D,D]
  const float* l1b     = (const float*)d_in[13];  // [D]
  float* out = (float*)d_out;                     // pred_r[B] ++ pred_n[B]

  // workspace layout (floats, all regions 16B aligned)
  float* w = (float*)d_ws;
  const size_t OFF_FLOW  = 0;                         // N*D        = 12,800,000
  const size_t OFF_SCORE = 12800000;                  // N          = 100,000
  const size_t OFF_PART  = 12900000;                  // 256*10*128 = 327,680
  const size_t OFF_CHAR  = 13227680;                  // 10*128
  const size_t OFF_LAT   = 13228960;                  // 2*10*128
  const size_t OFF_WEFF  = 13231520;                  // 128*128
  const size_t OFF_CONST = 13247904;                  // 8
  const size_t OFF_AGG   = 13247912;                  // B*D = 4,194,304 (reused as out)

  const int G = 256, chunk = (N_NODES + G - 1) / G;   // 391

  k_gemm_flow   <<<N_NODES / (16 * T1), 256, 0, stream>>>(feature, wemb, bemb, w + OFF_FLOW);
  k_char_partial<<<G, 128, 0, stream>>>(w + OFF_FLOW, adj, w + OFF_PART, chunk);
  k_char_reduce <<<M_CH, 128, 0, stream>>>(w + OFF_PART, w + OFF_CHAR, G);
  k_tables      <<<1, 256, 0, stream>>>(w + OFF_CHAR, wchar, attw, l1w,
                                        w + OFF_LAT, w + OFF_WEFF, w + OFF_CONST);
  k_score       <<<N_NODES / 8, 256, 0, stream>>>(w + OFF_FLOW, attw, adj,
                                                  w + OFF_CONST, attb, w + OFF_SCORE);
  k_attend      <<<B_BATCH / 8, 256, 0, stream>>>(w + OFF_FLOW, w + OFF_SCORE, adj,
                                                  w + OFF_CHAR, history, w + OFF_AGG);
  k_gemm_out    <<<B_BATCH / (16 * T2), 256, 0, stream>>>(w + OFF_AGG, w + OFF_WEFF, l1b);
  k_pred        <<<B_BATCH / 8, 256, 0, stream>>>(w + OFF_AGG, w + OFF_LAT,
                                                  cat_r, cat_n, pa_lvl, out);
}